// ScaledDotProductAttentionFused_74191265071203
// MI455X (gfx1250) — compile-verified
//
#include <hip/hip_runtime.h>

typedef float v2f __attribute__((ext_vector_type(2)));
typedef float v8f __attribute__((ext_vector_type(8)));

#define T_    2048
#define H_    8
#define M_    4
#define D_    128
#define WIN_  128
#define SCALE_ 0.08838834764831845f
#define MINIT_ -1.0e30f
#define MASKV_ -2.0e30f

// LDS row strides in dwords (keep 8B fragment alignment: all even; KSTR mult of 4 for float4 stores)
#define KSTR 132   // K tile: 16 rows x 128 cols
#define VSTR 18    // V^T tile: 128 rows x 16 cols
#define PSTR 18    // P scratch per wave: 16 rows x 16 cols

__global__ __launch_bounds__(128)
void attn_swa_sink_kernel(const float* __restrict__ Q, const float* __restrict__ K,
                          const float* __restrict__ V, const float* __restrict__ S,
                          float* __restrict__ out) {
  __shared__ float lKt[16 * KSTR];        // K tile, row-major [key][dim]
  __shared__ float lVt[128 * VSTR];       // V tile transposed [dim][key]
  __shared__ float lP [4 * 16 * PSTR];    // per-wave P scratch [wave][q][k]

  const int qt   = blockIdx.x;            // query tile index
  const int h    = blockIdx.y;
  const int q0   = qt * 16;
  const int tid  = threadIdx.x;
  const int wave = tid >> 5;              // q_mult index m
  const int lane = tid & 31;
  const int l16  = lane & 15;
  const int half = lane >> 4;

  // ---- Q A-fragments (16x4 fp32 tiles), kept in registers ----
  // A layout: lanes 0-15 hold row=lane, K={0,1} in vgpr{0,1}; lanes 16-31 hold K={2,3}.
  v2f a[32];
  {
    const float* qrow = Q + ((size_t)(q0 + l16) * (H_ * M_) + h * M_ + wave) * D_;
#pragma unroll
    for (int j = 0; j < 32; ++j) {
      a[j] = *(const v2f*)(qrow + 4 * j + 2 * half);
    }
  }

  // ---- accumulators ----
  v8f o[8];
  float mrow[8], lrow[8];
#pragma unroll
  for (int t = 0; t < 8; ++t)
#pragma unroll
    for (int v = 0; v < 8; ++v) o[t][v] = 0.0f;
#pragma unroll
  for (int v = 0; v < 8; ++v) { mrow[v] = MINIT_; lrow[v] = 0.0f; }

  float* lPw = lP + wave * 16 * PSTR;

  int kt_lo = qt - 8;
  if (kt_lo < 0) kt_lo = 0;

  for (int kt = kt_lo; kt <= qt; ++kt) {
    const int k0 = kt * 16;

    __syncthreads();  // previous tile's LDS reads complete before restaging

    // ---- cooperative staging: K tile row-major, V tile transposed ----
    {
      const int r  = tid >> 3;            // key row 0..15
      const int c0 = (tid & 7) * 16;      // dim chunk
      const float* kr = K + ((size_t)(k0 + r) * H_ + h) * D_ + c0;
      const float* vr = V + ((size_t)(k0 + r) * H_ + h) * D_ + c0;
#pragma unroll
      for (int u = 0; u < 4; ++u) {
        float4 kv = *(const float4*)(kr + 4 * u);
        *(float4*)(&lKt[r * KSTR + c0 + 4 * u]) = kv;
        float4 vv = *(const float4*)(vr + 4 * u);
        lVt[(c0 + 4 * u + 0) * VSTR + r] = vv.x;
        lVt[(c0 + 4 * u + 1) * VSTR + r] = vv.y;
        lVt[(c0 + 4 * u + 2) * VSTR + r] = vv.z;
        lVt[(c0 + 4 * u + 3) * VSTR + r] = vv.w;
      }
      if (kt < qt) {  // prefetch next key tile into cache (global_prefetch_b8)
        __builtin_prefetch(kr + (size_t)16 * H_ * D_, 0, 0);
        __builtin_prefetch(vr + (size_t)16 * H_ * D_, 0, 0);
      }
    }

    __syncthreads();

    // ---- S = Q x K^T via 32x V_WMMA_F32_16X16X4_F32 ----
    v8f s;
#pragma unroll
    for (int v = 0; v < 8; ++v) s[v] = 0.0f;
    {
      const float* kb = &lKt[l16 * KSTR + 2 * half];
#pragma unroll
      for (int j = 0; j < 32; ++j) {
        v2f b = *(const v2f*)(kb + 4 * j);
        s = __builtin_amdgcn_wmma_f32_16x16x4_f32(false, a[j], false, b,
                                                  (short)0, s, false, false);
      }
    }

    // ---- scale + sliding-window causal mask + online softmax ----
    float al[8];
#pragma unroll
    for (int v = 0; v < 8; ++v) {
      const int qg = q0 + v + 8 * half;   // C layout: vgpr v holds row v + 8*half
      const int kg = k0 + l16;            // column = lane % 16
      float val = s[v] * SCALE_;
      if (kg > qg || kg <= qg - WIN_) val = MASKV_;
      float r = val;                      // row-max over the 16-lane half
      r = fmaxf(r, __shfl_xor(r, 1, 32));
      r = fmaxf(r, __shfl_xor(r, 2, 32));
      r = fmaxf(r, __shfl_xor(r, 4, 32));
      r = fmaxf(r, __shfl_xor(r, 8, 32));
      const float mn = fmaxf(mrow[v], r);
      const float alpha = __expf(mrow[v] - mn);
      mrow[v] = mn;
      const float p = __expf(val - mn);
      float rs = p;                       // row-sum
      rs += __shfl_xor(rs, 1, 32);
      rs += __shfl_xor(rs, 2, 32);
      rs += __shfl_xor(rs, 4, 32);
      rs += __shfl_xor(rs, 8, 32);
      lrow[v] = lrow[v] * alpha + rs;
      al[v] = alpha;
      s[v] = p;
    }

    // rescale running O by alpha
#pragma unroll
    for (int t = 0; t < 8; ++t)
#pragma unroll
      for (int v = 0; v < 8; ++v) o[t][v] *= al[v];

    // ---- C-layout -> A-layout transpose of P through per-wave LDS ----
#pragma unroll
    for (int v = 0; v < 8; ++v)
      lPw[(v + 8 * half) * PSTR + l16] = s[v];

    __syncthreads();  // uniform across waves; also orders P store->load

    v2f pa[4];
#pragma unroll
    for (int c = 0; c < 4; ++c)
      pa[c] = *(const v2f*)(&lPw[l16 * PSTR + 4 * c + 2 * half]);

    // ---- O += P x V : 8 d-tiles x 4 WMMAs ----
#pragma unroll
    for (int t = 0; t < 8; ++t) {
#pragma unroll
      for (int c = 0; c < 4; ++c) {
        v2f bv = *(const v2f*)(&lVt[(16 * t + l16) * VSTR + 4 * c + 2 * half]);
        o[t] = __builtin_amdgcn_wmma_f32_16x16x4_f32(false, pa[c], false, bv,
                                                     (short)0, o[t], false, false);
      }
    }
  }

  // ---- fold in sink logit, normalize, store ----
  {
    const float sv = S[h * M_ + wave];
    float al[8];
#pragma unroll
    for (int v = 0; v < 8; ++v) {
      const float mn = fmaxf(mrow[v], sv);
      const float alpha = __expf(mrow[v] - mn);
      const float li = lrow[v] * alpha + __expf(sv - mn);
      al[v] = alpha / li;
    }
#pragma unroll
    for (int t = 0; t < 8; ++t) {
#pragma unroll
      for (int v = 0; v < 8; ++v) {
        const int qg = q0 + v + 8 * half;
        const int d  = 16 * t + l16;
        out[(size_t)qg * (H_ * M_ * D_) + (h * M_ + wave) * D_ + d] = o[t][v] * al[v];
      }
    }
  }
}

extern "C" void kernel_launch(void* const* d_in, const int* in_sizes, int n_in,
                              void* d_out, int out_size, void* d_ws, size_t ws_size,
                              hipStream_t stream) {
  (void)in_sizes; (void)n_in; (void)out_size; (void)d_ws; (void)ws_size;
  const float* Q = (const float*)d_in[0];
  const float* K = (const float*)d_in[1];
  const float* V = (const float*)d_in[2];
  const float* S = (const float*)d_in[3];
  float* out = (float*)d_out;
  dim3 grid(T_ / 16, H_);
  dim3 block(128);
  attn_swa_sink_kernel<<<grid, block, 0, stream>>>(Q, K, V, S, out);
}